// GCN_Net_68659347194092
// MI455X (gfx1250) — compile-verified
//
#include <hip/hip_runtime.h>
#include <math.h>

typedef __attribute__((ext_vector_type(2))) float v2f;
typedef __attribute__((ext_vector_type(4))) float v4f;
typedef __attribute__((ext_vector_type(8))) float v8f;

#define NN 10000
#define EE 160000
#define DD 512

// ---------------------------------------------------------------------------
// Elementwise: out = x + cos(t[n]*tw[d] + tb[d])
// ---------------------------------------------------------------------------
__global__ void time_encode_kernel(const float* __restrict__ x,
                                   const float* __restrict__ t,
                                   const float* __restrict__ tw,
                                   const float* __restrict__ tb,
                                   float* __restrict__ out) {
    int idx = blockIdx.x * blockDim.x + threadIdx.x;   // [0, N*D)
    int n = idx >> 9;
    int d = idx & (DD - 1);
    out[idx] = x[idx] + cosf(t[n] * tw[d] + tb[d]);
}

// ---------------------------------------------------------------------------
// CSR build: degree histogram -> scan (+dinv) -> bucket fill
// ---------------------------------------------------------------------------
__global__ void zero_int_kernel(int* __restrict__ p, int n) {
    int i = blockIdx.x * blockDim.x + threadIdx.x;
    if (i < n) p[i] = 0;
}

__global__ void deg_count_kernel(const int* __restrict__ edges, // [2,E]
                                 int* __restrict__ deg) {
    int e = blockIdx.x * blockDim.x + threadIdx.x;
    if (e < EE) atomicAdd(&deg[edges[EE + e]], 1);              // dst = row 1
}

// single workgroup, 1024 threads: exclusive scan of deg -> rowstart,
// rowstart[NN] = E, and dinv[n] = rsqrt(deg[n]+1)
__global__ void __launch_bounds__(1024)
scan_dinv_kernel(const int* __restrict__ deg,
                 int* __restrict__ rowstart,
                 float* __restrict__ dinv) {
    __shared__ int partial[1024];
    const int tid = threadIdx.x;
    const int PER = 10;                       // 1024*10 >= 10000
    int local[PER];
    int base = tid * PER;
    int s = 0;
    #pragma unroll
    for (int i = 0; i < PER; ++i) {
        int idx = base + i;
        int v = (idx < NN) ? deg[idx] : 0;
        local[i] = s;                          // thread-local exclusive
        s += v;
    }
    partial[tid] = s;
    __syncthreads();
    // inclusive Hillis-Steele scan over the 1024 partials
    for (int off = 1; off < 1024; off <<= 1) {
        int v = (tid >= off) ? partial[tid - off] : 0;
        __syncthreads();
        partial[tid] += v;
        __syncthreads();
    }
    int offset = (tid == 0) ? 0 : partial[tid - 1];
    #pragma unroll
    for (int i = 0; i < PER; ++i) {
        int idx = base + i;
        if (idx < NN) rowstart[idx] = offset + local[i];
    }
    if (tid == 1023) rowstart[NN] = partial[1023];   // == E
    for (int i = tid; i < NN; i += 1024)
        dinv[i] = rsqrtf((float)deg[i] + 1.0f);      // +1 self loop
}

__global__ void fill_csr_kernel(const int* __restrict__ edges,  // [2,E]
                                const int* __restrict__ rowstart,
                                int* __restrict__ cursor,
                                int* __restrict__ csr_src) {
    int e = blockIdx.x * blockDim.x + threadIdx.x;
    if (e >= EE) return;
    int src = edges[e];
    int dst = edges[EE + e];
    int pos = atomicAdd(&cursor[dst], 1);
    csr_src[rowstart[dst] + pos] = src;
}

// ---------------------------------------------------------------------------
// FP32 WMMA GEMM: C[10000 x 512] = A[10000 x 512] @ W[512 x 512]
// V_WMMA_F32_16X16X4_F32, wave computes 16x64 tile (4 accumulators).
// block = 128 threads (4 waves); grid = (625, 2).
// ---------------------------------------------------------------------------
__global__ void __launch_bounds__(128)
gemm_f32_wmma_kernel(const float* __restrict__ A,
                     const float* __restrict__ W,
                     float* __restrict__ C) {
    const int lane  = threadIdx.x & 31;
    const int wave  = threadIdx.x >> 5;
    const int half  = lane >> 4;        // K pair (A/B) / M half (C)
    const int r16   = lane & 15;        // A row within tile / C-D column
    const int m0    = blockIdx.x * 16;
    const int n0    = blockIdx.y * 256 + wave * 64;

    const float* Arow = A + (size_t)(m0 + r16) * DD;

    v8f acc0 = {}, acc1 = {}, acc2 = {}, acc3 = {};

    #pragma unroll 4
    for (int k = 0; k < DD; k += 4) {
        const int ka = k + 2 * half;
        v2f a = *(const v2f*)(Arow + ka);

        const float* Wk0 = W + (size_t)ka * DD;
        const float* Wk1 = Wk0 + DD;
        const int nc = n0 + r16;
        v2f b0 = { Wk0[nc +  0], Wk1[nc +  0] };
        v2f b1 = { Wk0[nc + 16], Wk1[nc + 16] };
        v2f b2 = { Wk0[nc + 32], Wk1[nc + 32] };
        v2f b3 = { Wk0[nc + 48], Wk1[nc + 48] };

        acc0 = __builtin_amdgcn_wmma_f32_16x16x4_f32(false, a, false, b0,
                                                     (short)0, acc0, false, false);
        acc1 = __builtin_amdgcn_wmma_f32_16x16x4_f32(false, a, false, b1,
                                                     (short)0, acc1, false, false);
        acc2 = __builtin_amdgcn_wmma_f32_16x16x4_f32(false, a, false, b2,
                                                     (short)0, acc2, false, false);
        acc3 = __builtin_amdgcn_wmma_f32_16x16x4_f32(false, a, false, b3,
                                                     (short)0, acc3, false, false);
    }

    #pragma unroll
    for (int r = 0; r < 8; ++r) {
        float* crow = C + (size_t)(m0 + 8 * half + r) * DD + n0 + r16;
        crow[0]  = acc0[r];
        crow[16] = acc1[r];
        crow[32] = acc2[r];
        crow[48] = acc3[r];
    }
}

// ---------------------------------------------------------------------------
// Gather conv: Out[n] = b + H[n]*dinv[n]^2 + sum_{s in in(n)} H[s]*dinv[s]*dinv[n]
// One 128-thread block per dst node, float4 per lane, zero atomics.
// ---------------------------------------------------------------------------
__global__ void __launch_bounds__(128)
gather_conv_kernel(const float* __restrict__ H,
                   const int* __restrict__ csr_src,
                   const int* __restrict__ rowstart,
                   const float* __restrict__ dinv,
                   const float* __restrict__ bias,
                   float* __restrict__ Out) {
    const int n = blockIdx.x;
    const int d = threadIdx.x * 4;
    const int beg = rowstart[n];
    const int end = rowstart[n + 1];
    const float dn = dinv[n];

    v4f acc = *(const v4f*)(H + (size_t)n * DD + d);
    v4f bv  = *(const v4f*)(bias + d);
    acc = acc * (dn * dn) + bv;

    for (int i = beg; i < end; ++i) {
        int s = csr_src[i];
        if (i + 1 < end) {                               // cover L2 latency
            int s2 = csr_src[i + 1];
            __builtin_prefetch(H + (size_t)s2 * DD + d, 0, 3);
        }
        float c = dinv[s] * dn;
        v4f hv = *(const v4f*)(H + (size_t)s * DD + d);
        acc += hv * c;
    }
    *(v4f*)(Out + (size_t)n * DD + d) = acc;
}

// ---------------------------------------------------------------------------
// logits accumulation: one wave per node, coalesced, shuffle reduction
// ---------------------------------------------------------------------------
__global__ void fc_accum_kernel(const float* __restrict__ Hb,
                                const float* __restrict__ fc_w,  // [1024,2]
                                const float* __restrict__ fc_b,
                                float* __restrict__ logits,
                                int k_off, int init) {
    int gid  = blockIdx.x * blockDim.x + threadIdx.x;
    int node = gid >> 5;
    int lane = gid & 31;
    if (node >= NN) return;
    const float* h = Hb + (size_t)node * DD;
    float l0 = 0.0f, l1 = 0.0f;
    #pragma unroll 4
    for (int k = lane; k < DD; k += 32) {
        float hv = h[k];
        l0 += hv * fc_w[(k_off + k) * 2 + 0];
        l1 += hv * fc_w[(k_off + k) * 2 + 1];
    }
    #pragma unroll
    for (int off = 16; off > 0; off >>= 1) {
        l0 += __shfl_down(l0, off, 32);
        l1 += __shfl_down(l1, off, 32);
    }
    if (lane == 0) {
        if (init) { l0 += fc_b[0]; l1 += fc_b[1]; }
        else      { l0 += logits[2 * node + 0]; l1 += logits[2 * node + 1]; }
        logits[2 * node + 0] = l0;
        logits[2 * node + 1] = l1;
    }
}

__global__ void logsoftmax_kernel(const float* __restrict__ logits,
                                  float* __restrict__ out) {
    int n = blockIdx.x * blockDim.x + threadIdx.x;
    if (n >= NN) return;
    float a = logits[2 * n + 0], b = logits[2 * n + 1];
    float m = fmaxf(a, b);
    float lse = m + logf(expf(a - m) + expf(b - m));
    out[2 * n + 0] = a - lse;
    out[2 * n + 1] = b - lse;
}

// ---------------------------------------------------------------------------
// Host-side orchestration
// ---------------------------------------------------------------------------
static void run_branch(const float* x, const float* t,
                       const float* tw, const float* tb,
                       const float* W1, const float* b1,
                       const float* W2, const float* b2,
                       const int* edges,
                       const float* fc_w, const float* fc_b,
                       float* B0, float* B1, float* B2,
                       float* logits, float* dinv,
                       int* deg, int* rowstart, int* cursor, int* csr_src,
                       int k_off, int init, hipStream_t stream) {
    const int ND = NN * DD;
    // CSR build + dinv
    zero_int_kernel<<<(NN + 255) / 256, 256, 0, stream>>>(deg, NN);
    zero_int_kernel<<<(NN + 255) / 256, 256, 0, stream>>>(cursor, NN);
    deg_count_kernel<<<(EE + 255) / 256, 256, 0, stream>>>(edges, deg);
    scan_dinv_kernel<<<1, 1024, 0, stream>>>(deg, rowstart, dinv);
    fill_csr_kernel<<<(EE + 255) / 256, 256, 0, stream>>>(edges, rowstart,
                                                          cursor, csr_src);
    // x + time encoding -> B0
    time_encode_kernel<<<ND / 256, 256, 0, stream>>>(x, t, tw, tb, B0);
    // conv1: H = B0 @ W1 -> B1 ; gather -> B2
    gemm_f32_wmma_kernel<<<dim3(NN / 16, 2), 128, 0, stream>>>(B0, W1, B1);
    gather_conv_kernel<<<NN, 128, 0, stream>>>(B1, csr_src, rowstart, dinv,
                                               b1, B2);
    // conv2: H = B2 @ W2 -> B0 ; gather -> B1
    gemm_f32_wmma_kernel<<<dim3(NN / 16, 2), 128, 0, stream>>>(B2, W2, B0);
    gather_conv_kernel<<<NN, 128, 0, stream>>>(B0, csr_src, rowstart, dinv,
                                               b2, B1);
    // logits accumulation (wave per node)
    fc_accum_kernel<<<(NN * 32 + 255) / 256, 256, 0, stream>>>(B1, fc_w, fc_b,
                                                               logits, k_off,
                                                               init);
}

extern "C" void kernel_launch(void* const* d_in, const int* in_sizes, int n_in,
                              void* d_out, int out_size, void* d_ws, size_t ws_size,
                              hipStream_t stream) {
    const float* x        = (const float*)d_in[0];
    const int*   e_in     = (const int*)  d_in[1];
    const int*   e_out    = (const int*)  d_in[2];
    const float* t        = (const float*)d_in[3];
    const float* te_w_in  = (const float*)d_in[4];
    const float* te_b_in  = (const float*)d_in[5];
    const float* W1_in    = (const float*)d_in[6];
    const float* b1_in    = (const float*)d_in[7];
    const float* W2_in    = (const float*)d_in[8];
    const float* b2_in    = (const float*)d_in[9];
    const float* te_w_out = (const float*)d_in[10];
    const float* te_b_out = (const float*)d_in[11];
    const float* W1_out   = (const float*)d_in[12];
    const float* b1_out   = (const float*)d_in[13];
    const float* W2_out   = (const float*)d_in[14];
    const float* b2_out   = (const float*)d_in[15];
    const float* fc_w     = (const float*)d_in[16];
    const float* fc_b     = (const float*)d_in[17];

    float* B0     = (float*)d_ws;
    float* B1     = B0 + (size_t)NN * DD;
    float* B2     = B1 + (size_t)NN * DD;
    float* logits = B2 + (size_t)NN * DD;
    float* dinv   = logits + (size_t)NN * 2;
    int*   deg      = (int*)(dinv + NN);
    int*   rowstart = deg + NN;
    int*   cursor   = rowstart + (NN + 1);
    int*   csr_src  = cursor + NN;

    run_branch(x, t, te_w_in, te_b_in, W1_in, b1_in, W2_in, b2_in,
               e_in, fc_w, fc_b, B0, B1, B2, logits, dinv,
               deg, rowstart, cursor, csr_src,
               /*k_off=*/0, /*init=*/1, stream);
    run_branch(x, t, te_w_out, te_b_out, W1_out, b1_out, W2_out, b2_out,
               e_out, fc_w, fc_b, B0, B1, B2, logits, dinv,
               deg, rowstart, cursor, csr_src,
               /*k_off=*/DD, /*init=*/0, stream);

    logsoftmax_kernel<<<(NN + 255) / 256, 256, 0, stream>>>(logits,
                                                            (float*)d_out);
}